// Net_75995151335951
// MI455X (gfx1250) — compile-verified
//
#include <hip/hip_runtime.h>
#include <math.h>

typedef __attribute__((ext_vector_type(2))) float v2f;
typedef __attribute__((ext_vector_type(8))) float v8f;

#define HEADS 4
#define NEG_SLOPE 0.2f
#define KB 64              // K-chunk staged in LDS per iteration
#define LSTR 68            // LDS row stride (floats): stride%64banks==4 -> conflict-free reads

// ---------------------------------------------------------------------------
// fp32 GEMM: Y[M,Ncols] = X[M,K] @ W[K,Ncols] via V_WMMA_F32_16X16X4_F32.
// Block = 128 threads (4 waves) computes a 32x64 tile of Y.
// Per KB=64 chunk: stage A (32xKB) and B (KBx64, transposed) into LDS with
// zero padding for ragged M/K/Ncols, then a branch-free fully-unrolled loop
// of ds_load_b64 + wmma. EXEC stays all-ones across every WMMA.
// ---------------------------------------------------------------------------
__global__ __launch_bounds__(128)
void gat_gemm_wmma(const float* __restrict__ X, const float* __restrict__ W,
                   float* __restrict__ Y, int M, int K, int Ncols) {
    __shared__ float ldsA[32 * LSTR];
    __shared__ float ldsB[64 * LSTR];
    const int tid   = threadIdx.x;
    const int tileM = blockIdx.y * 32;
    const int col0  = blockIdx.x * 64;

    const int lane     = tid & 31;
    const int wave     = tid >> 5;
    const int nlane    = lane & 15;
    const int khalf    = (lane >> 4) << 1;      // 0 (lanes 0-15) or 2 (lanes 16-31)
    const int colLocal = wave * 16 + nlane;
    const int col      = col0 + colLocal;
    const bool colok   = (col < Ncols);

    v8f acc0 = {}, acc1 = {};
    const int nChunks = (K + KB - 1) / KB;

    for (int ki = 0; ki < nChunks; ++ki) {
        const int k0 = ki * KB;
        const bool more = (ki + 1 < nChunks);

        // Stage A: 32 rows x KB cols (zero-padded). Coalesced global reads.
        for (int idx = tid; idx < 32 * KB; idx += 128) {
            int r = idx >> 6, c = idx & 63;
            int gr = tileM + r, gc = k0 + c;
            bool ok = (gr < M) & (gc < K);
            ldsA[r * LSTR + c] = ok ? X[(long)gr * K + gc] : 0.0f;
            if (more && ok && ((c & 15) == 0) && (gc + KB) < K)
                __builtin_prefetch(&X[(long)gr * K + gc + KB], 0, 0);   // global_prefetch_b8
        }
        // Stage B transposed: ldsB[col][k] so each lane's K-pair is contiguous.
        for (int idx = tid; idx < KB * 64; idx += 128) {
            int r = idx >> 6, c = idx & 63;     // r = k offset, c = col offset
            int gk = k0 + r, gc = col0 + c;
            bool ok = (gk < K) & (gc < Ncols);
            ldsB[c * LSTR + r] = ok ? W[(long)gk * Ncols + gc] : 0.0f;
            if (more && ok && ((c & 15) == 0) && (gk + KB) < K)
                __builtin_prefetch(&W[(long)(gk + KB) * Ncols + gc], 0, 0);
        }
        __syncthreads();

        const float* pa0 = &ldsA[nlane * LSTR + khalf];
        const float* pa1 = &ldsA[(16 + nlane) * LSTR + khalf];
        const float* pb  = &ldsB[colLocal * LSTR + khalf];
#pragma unroll
        for (int kk = 0; kk < KB; kk += 4) {
            // A 16x4 fp32 layout: VGPR0=K(kk+khalf), VGPR1=K(kk+khalf+1), M=lane&15
            v2f a0 = *(const v2f*)(pa0 + kk);
            v2f a1 = *(const v2f*)(pa1 + kk);
            // B 4x16 fp32 layout mirrored: row striped across lanes, halves hold K pairs
            v2f b  = *(const v2f*)(pb + kk);
            acc0 = __builtin_amdgcn_wmma_f32_16x16x4_f32(
                false, a0, false, b, (short)0, acc0, false, false);
            acc1 = __builtin_amdgcn_wmma_f32_16x16x4_f32(
                false, a1, false, b, (short)0, acc1, false, false);
        }
        __syncthreads();
    }

    // D 16x16 f32 layout: VGPR i -> M=i (lanes 0-15) / 8+i (lanes 16-31), N=lane&15
    const int rb = ((lane >> 4) << 3);
#pragma unroll
    for (int i = 0; i < 8; ++i) {
        int r0 = tileM + rb + i;
        int r1 = r0 + 16;
        if (colok && r0 < M) Y[(long)r0 * Ncols + col] = acc0[i];
        if (colok && r1 < M) Y[(long)r1 * Ncols + col] = acc1[i];
    }
}

// ---------------------------------------------------------------------------
// Attention logits: alS[n,h] = sum_c H[n,h,c]*aS[h,c]; alD likewise.
// ---------------------------------------------------------------------------
__global__ void gat_al(const float* __restrict__ Hb, const float* __restrict__ aS,
                       const float* __restrict__ aD, float* __restrict__ alS,
                       float* __restrict__ alD, int Nn, int Cout) {
    int t = blockIdx.x * blockDim.x + threadIdx.x;
    if (t >= Nn * HEADS) return;
    int n = t >> 2, h = t & 3;
    const float* hp = Hb + (long)n * HEADS * Cout + h * Cout;
    const float* as = aS + h * Cout;
    const float* ad = aD + h * Cout;
    float ss = 0.f, sd = 0.f;
    for (int c = 0; c < Cout; ++c) { float v = hp[c]; ss += v * as[c]; sd += v * ad[c]; }
    alS[t] = ss; alD[t] = sd;
}

__global__ void gat_fill(float* __restrict__ p, float v, long n) {
    long i = (long)blockIdx.x * blockDim.x + threadIdx.x;
    if (i < n) p[i] = v;
}

__device__ __forceinline__ void edge_sd(const int* __restrict__ ei, int E, int e,
                                        int& s, int& d) {
    if (e < E) { s = ei[e]; d = ei[E + e]; } else { s = e - E; d = s; }  // self loops
}

__device__ __forceinline__ float leaky(float x) {
    return x > 0.f ? x : NEG_SLOPE * x;
}

__device__ __forceinline__ void atomicMaxF(float* addr, float val) {
    unsigned int* ua = (unsigned int*)addr;
    unsigned int old = *ua;
    while (__uint_as_float(old) < val) {
        unsigned int assumed = old;
        old = atomicCAS(ua, assumed, __float_as_uint(val));
        if (old == assumed) break;
    }
}

// Segment max of leaky(alS[src]+alD[dst]) per (dst, head).
__global__ void gat_edge_max(const int* __restrict__ ei, int E, int Etot,
                             const float* __restrict__ alS, const float* __restrict__ alD,
                             float* __restrict__ m) {
    long t = (long)blockIdx.x * blockDim.x + threadIdx.x;
    if (t >= (long)Etot * HEADS) return;
    int e = (int)(t >> 2), h = (int)(t & 3);
    int s, d; edge_sd(ei, E, e, s, d);
    float lr = leaky(alS[s * HEADS + h] + alD[d * HEADS + h]);
    atomicMaxF(&m[d * HEADS + h], lr);
}

// ex = exp(e - m[dst]); denom[dst,h] += ex; store ex per (edge,head).
__global__ void gat_edge_exp(const int* __restrict__ ei, int E, int Etot,
                             const float* __restrict__ alS, const float* __restrict__ alD,
                             const float* __restrict__ m, float* __restrict__ EX,
                             float* __restrict__ den) {
    long t = (long)blockIdx.x * blockDim.x + threadIdx.x;
    if (t >= (long)Etot * HEADS) return;
    int e = (int)(t >> 2), h = (int)(t & 3);
    int s, d; edge_sd(ei, E, e, s, d);
    float lr = leaky(alS[s * HEADS + h] + alD[d * HEADS + h]);
    float ex = expf(lr - m[d * HEADS + h]);
    EX[t] = ex;
    atomicAdd(&den[d * HEADS + h], ex);
}

// AGG[dst,h,c] += H[src,h,c] * (EX[e,h]/den[dst,h]). One block per edge.
__global__ void gat_aggregate(const int* __restrict__ ei, int E,
                              const float* __restrict__ Hb, const float* __restrict__ EX,
                              const float* __restrict__ den, float* __restrict__ AGG,
                              int Cout) {
    int e = blockIdx.x;
    int s, d; edge_sd(ei, E, e, s, d);
    const int HC = HEADS * Cout;
    for (int j = threadIdx.x; j < HC; j += blockDim.x) {
        int h = j / Cout;
        float alpha = EX[e * HEADS + h] / den[d * HEADS + h];
        atomicAdd(&AGG[(long)d * HC + j], Hb[(long)s * HC + j] * alpha);
    }
}

// Layers 1,2: X' = relu(AGG + R + b), concat layout [N, 256].
__global__ void gat_finalize_relu(const float* __restrict__ AGG, const float* __restrict__ R,
                                  const float* __restrict__ b, float* __restrict__ Xout,
                                  int Nn) {
    long t = (long)blockIdx.x * blockDim.x + threadIdx.x;
    if (t >= (long)Nn * 256) return;
    int j = (int)(t & 255);
    float v = AGG[t] + R[t] + b[j];
    Xout[t] = v > 0.f ? v : 0.f;
}

// Layer 3: out = mean_h(AGG[n,h,:]) + R + b, [N, 121].
__global__ void gat_finalize_mean(const float* __restrict__ AGG, const float* __restrict__ R,
                                  const float* __restrict__ b, float* __restrict__ out,
                                  int Nn) {
    long t = (long)blockIdx.x * blockDim.x + threadIdx.x;
    if (t >= (long)Nn * 121) return;
    int n = (int)(t / 121), c = (int)(t - (long)n * 121);
    const float* a = AGG + (long)n * 484;
    out[t] = 0.25f * (a[c] + a[121 + c] + a[242 + c] + a[363 + c]) + R[t] + b[c];
}

// ---------------------------------------------------------------------------
static void gat_layer(const float* Xin, int K, int Cout, bool mean_out,
                      const float* W, const float* aS, const float* aD,
                      const float* resW, const float* b, float* Xout,
                      float* Hb, float* R, float* AGG, float* alS, float* alD,
                      float* mMax, float* den, float* EX,
                      const int* ei, int Nn, int E, int Etot, hipStream_t stream) {
    const int HC = HEADS * Cout;
    const int RC = mean_out ? Cout : HC;   // residual width (layer3 resW is [K,NC])

    dim3 gb((HC + 63) / 64, (Nn + 31) / 32);
    gat_gemm_wmma<<<gb, 128, 0, stream>>>(Xin, W, Hb, Nn, K, HC);
    dim3 gr((RC + 63) / 64, (Nn + 31) / 32);
    gat_gemm_wmma<<<gr, 128, 0, stream>>>(Xin, resW, R, Nn, K, RC);

    const int nh = Nn * HEADS;
    gat_al<<<(nh + 255) / 256, 256, 0, stream>>>(Hb, aS, aD, alS, alD, Nn, Cout);

    gat_fill<<<(nh + 255) / 256, 256, 0, stream>>>(mMax, -3.0e38f, nh);
    gat_fill<<<(nh + 255) / 256, 256, 0, stream>>>(den, 0.0f, nh);
    long aggn = (long)Nn * HC;
    gat_fill<<<(unsigned)((aggn + 255) / 256), 256, 0, stream>>>(AGG, 0.0f, aggn);

    long eh = (long)Etot * HEADS;
    gat_edge_max<<<(unsigned)((eh + 255) / 256), 256, 0, stream>>>(ei, E, Etot, alS, alD, mMax);
    gat_edge_exp<<<(unsigned)((eh + 255) / 256), 256, 0, stream>>>(ei, E, Etot, alS, alD, mMax, EX, den);
    gat_aggregate<<<Etot, 256, 0, stream>>>(ei, E, Hb, EX, den, AGG, Cout);

    if (mean_out)
        gat_finalize_mean<<<(unsigned)(((long)Nn * 121 + 255) / 256), 256, 0, stream>>>(AGG, R, b, Xout, Nn);
    else
        gat_finalize_relu<<<(unsigned)(((long)Nn * 256 + 255) / 256), 256, 0, stream>>>(AGG, R, b, Xout, Nn);
}

extern "C" void kernel_launch(void* const* d_in, const int* in_sizes, int n_in,
                              void* d_out, int out_size, void* d_ws, size_t ws_size,
                              hipStream_t stream) {
    const float* x0   = (const float*)d_in[0];
    const int*   ei   = (const int*)d_in[1];     // [2,E] flat: src then dst
    const float* W1   = (const float*)d_in[2];
    const float* as1  = (const float*)d_in[3];
    const float* ad1  = (const float*)d_in[4];
    const float* res1 = (const float*)d_in[5];
    const float* b1   = (const float*)d_in[6];
    const float* W2   = (const float*)d_in[7];
    const float* as2  = (const float*)d_in[8];
    const float* ad2  = (const float*)d_in[9];
    const float* res2 = (const float*)d_in[10];
    const float* b2   = (const float*)d_in[11];
    const float* W3   = (const float*)d_in[12];
    const float* as3  = (const float*)d_in[13];
    const float* ad3  = (const float*)d_in[14];
    const float* res3 = (const float*)d_in[15];
    const float* b3   = (const float*)d_in[16];

    const int Nn   = in_sizes[0] / 50;           // F_IN = 50
    const int E    = in_sizes[1] / 2;
    const int Etot = E + Nn;                     // + self loops

    float* ws  = (float*)d_ws;
    float* X    = ws;  ws += (size_t)Nn * 256;   // activations (reused across layers)
    float* Hb   = ws;  ws += (size_t)Nn * 484;   // h = x@W (max H*NC = 484 cols)
    float* R    = ws;  ws += (size_t)Nn * 256;   // residual x@resW
    float* AGG  = ws;  ws += (size_t)Nn * 484;   // scatter accumulator
    float* alS  = ws;  ws += (size_t)Nn * HEADS;
    float* alD  = ws;  ws += (size_t)Nn * HEADS;
    float* mMax = ws;  ws += (size_t)Nn * HEADS;
    float* den  = ws;  ws += (size_t)Nn * HEADS;
    float* EX   = ws;  ws += (size_t)Etot * HEADS;
    (void)ws_size; (void)n_in; (void)out_size;

    // Layer 1: F_IN=50 -> 4x64 concat, relu
    gat_layer(x0, 50, 64, false, W1, as1, ad1, res1, b1, X,
              Hb, R, AGG, alS, alD, mMax, den, EX, ei, Nn, E, Etot, stream);
    // Layer 2: 256 -> 4x64 concat, relu (X updated in place: finalize reads only AGG/R)
    gat_layer(X, 256, 64, false, W2, as2, ad2, res2, b2, X,
              Hb, R, AGG, alS, alD, mMax, den, EX, ei, Nn, E, Etot, stream);
    // Layer 3: 256 -> 121, mean over heads
    gat_layer(X, 256, 121, true, W3, as3, ad3, res3, b3, (float*)d_out,
              Hb, R, AGG, alS, alD, mMax, den, EX, ei, Nn, E, Etot, stream);
}